// MultiFrequencyModule_12524124635272
// MI455X (gfx1250) — compile-verified
//
#include <hip/hip_runtime.h>
#include <hip/hip_bf16.h>
#include <math.h>

// ---------------------------------------------------------------------------
// Dual-frequency GAT (MultiFrequencyModule) for gfx1250 (MI455X, wave32).
// Dense projection uses V_WMMA_F32_16X16X4_F32 (exact fp32 matrix core path).
// ---------------------------------------------------------------------------

typedef float v2f __attribute__((ext_vector_type(2)));
typedef float v8f __attribute__((ext_vector_type(8)));

#define F_IN 64
#define HC   128   // HEADS * C_OUT
#define COUT 64

// ---------------- elementwise prep: high = init - low; init accumulators ----
__global__ __launch_bounds__(256)
void prep_kernel(const float* __restrict__ init_f, const float* __restrict__ low_f,
                 float* __restrict__ xh,
                 float* __restrict__ ml, float* __restrict__ mh,
                 float* __restrict__ dl, float* __restrict__ dh,
                 const float* __restrict__ bias_l, const float* __restrict__ bias_h,
                 const float* __restrict__ alpha,
                 float* __restrict__ out, int n)
{
    int i = blockIdx.x * blockDim.x + threadIdx.x;
    int nf = n * F_IN;
    if (i < nf) {
        xh[i] = init_f[i] - low_f[i];
        float a = alpha[0];
        int c = i & (COUT - 1);           // out has n*64 elements, same count as nf
        out[i] = (1.0f - a) * bias_l[c] + a * bias_h[c];
    }
    if (i < n * 2) {
        ml[i] = -INFINITY;
        mh[i] = -INFINITY;
        dl[i] = 0.0f;
        dh[i] = 0.0f;
    }
}

// ---------------- WMMA fp32 GEMM: XP[n,128] = X[n,64] @ W[128,64]^T ---------
// One block (256 thr = 8 waves) handles one 16-row tile; wave w -> col tile w.
// A 16x4 f32 lane layout: lane<16 -> K={0,1}, lane>=16 -> K={2,3}; M = lane%16.
// B 4x16 f32 mirrors A with N = lane%16. D: VGPR v -> M = v + 8*(lane>=16).
__global__ __launch_bounds__(256)
void gemm_xp_kernel(const float* __restrict__ X, const float* __restrict__ W,
                    float* __restrict__ XP, int n)
{
    const int lane = threadIdx.x & 31;
    const int wave = threadIdx.x >> 5;        // 0..7 : column tile
    const int r    = lane & 15;
    const int hi   = lane >> 4;               // 0 or 1
    const int tm   = blockIdx.x;

    int row  = tm * 16 + r;
    int rowc = row < n ? row : (n - 1);       // clamp (keep EXEC all-ones for WMMA)
    const float* arow = X + (size_t)rowc * F_IN;
    const float* brow = W + (size_t)(wave * 16 + r) * F_IN;   // B[k][o] = W[o][k]

    v8f c = {0.f, 0.f, 0.f, 0.f, 0.f, 0.f, 0.f, 0.f};
#pragma unroll
    for (int kk = 0; kk < 16; ++kk) {
        int kb = kk * 4 + 2 * hi;
        v2f a; a.x = arow[kb]; a.y = arow[kb + 1];
        v2f b; b.x = brow[kb]; b.y = brow[kb + 1];
        c = __builtin_amdgcn_wmma_f32_16x16x4_f32(
                /*neg_a=*/false, a, /*neg_b=*/false, b,
                /*c_mod=*/(short)0, c, /*reuse_a=*/false, /*reuse_b=*/false);
    }

#pragma unroll
    for (int v = 0; v < 8; ++v) {
        int m = tm * 16 + v + 8 * hi;
        if (m < n)
            XP[(size_t)m * HC + wave * 16 + r] = c[v];
    }
}

// ---------------- per-node attention scores: a = <xp[n,h,:], att[h,:]> ------
__global__ __launch_bounds__(256)
void attn_score_kernel(const float* __restrict__ XP,
                       const float* __restrict__ att_src, const float* __restrict__ att_dst,
                       float* __restrict__ a_src, float* __restrict__ a_dst, int n)
{
    int i = blockIdx.x * blockDim.x + threadIdx.x;   // i indexes (node, head)
    if (i >= n * 2) return;
    int node = i >> 1, h = i & 1;
    const float* v  = XP + (size_t)node * HC + h * COUT;
    const float* as = att_src + h * COUT;
    const float* ad = att_dst + h * COUT;
    float ss = 0.f, sd = 0.f;
#pragma unroll 8
    for (int c = 0; c < COUT; ++c) {
        float x = v[c];
        ss += x * as[c];
        sd += x * ad[c];
    }
    a_src[i] = ss;
    a_dst[i] = sd;
}

// ---------------- edge helpers ----------------------------------------------
__device__ __forceinline__ float lrelu(float x) { return x > 0.f ? x : 0.2f * x; }

__device__ __forceinline__ void atomicMaxF(float* addr, float val)
{
    // monotone bit-pattern trick: works with init = -inf
    if (val >= 0.f) atomicMax((int*)addr, __float_as_int(val));
    else            atomicMin((unsigned int*)addr, __float_as_uint(val));
}

__device__ __forceinline__ void edge_endpoints(const int* ei, int E, int e, int& s, int& d)
{
    if (e < E) { s = ei[e]; d = ei[E + e]; }
    else       { s = d = e - E; }               // synthesized self-loop
}

// ---------------- pass 1: segment max over dst ------------------------------
__global__ __launch_bounds__(256)
void edge_max_kernel(const int* __restrict__ ei, int E, int n,
                     const float* __restrict__ asl, const float* __restrict__ adl,
                     const float* __restrict__ ash, const float* __restrict__ adh,
                     float* __restrict__ ml, float* __restrict__ mh)
{
    int e = blockIdx.x * blockDim.x + threadIdx.x;
    if (e >= E + n) return;
    int s, d; edge_endpoints(ei, E, e, s, d);
#pragma unroll
    for (int h = 0; h < 2; ++h) {
        atomicMaxF(&ml[d * 2 + h], lrelu(asl[s * 2 + h] + adl[d * 2 + h]));
        atomicMaxF(&mh[d * 2 + h], lrelu(ash[s * 2 + h] + adh[d * 2 + h]));
    }
}

// ---------------- pass 2: segment sum of exp(e - m) -------------------------
__global__ __launch_bounds__(256)
void edge_sum_kernel(const int* __restrict__ ei, int E, int n,
                     const float* __restrict__ asl, const float* __restrict__ adl,
                     const float* __restrict__ ash, const float* __restrict__ adh,
                     const float* __restrict__ ml, const float* __restrict__ mh,
                     float* __restrict__ dl, float* __restrict__ dh)
{
    int e = blockIdx.x * blockDim.x + threadIdx.x;
    if (e >= E + n) return;
    int s, d; edge_endpoints(ei, E, e, s, d);
#pragma unroll
    for (int h = 0; h < 2; ++h) {
        float el = lrelu(asl[s * 2 + h] + adl[d * 2 + h]);
        atomicAdd(&dl[d * 2 + h], expf(el - ml[d * 2 + h]));
        float eh = lrelu(ash[s * 2 + h] + adh[d * 2 + h]);
        atomicAdd(&dh[d * 2 + h], expf(eh - mh[d * 2 + h]));
    }
}

// ---------------- pass 3: fused weighted scatter (one wave per edge) --------
// out[d,c] += sum_h [ (1-a)/2 * attn_l * xp_l[s,h,c] + a/2 * attn_h * xp_h[s,h,c] ]
__global__ __launch_bounds__(256)
void edge_scatter_kernel(const int* __restrict__ ei, int E, int n,
                         const float* __restrict__ xpl, const float* __restrict__ xph,
                         const float* __restrict__ asl, const float* __restrict__ adl,
                         const float* __restrict__ ash, const float* __restrict__ adh,
                         const float* __restrict__ ml, const float* __restrict__ mh,
                         const float* __restrict__ dl, const float* __restrict__ dh,
                         const float* __restrict__ alpha, float* __restrict__ out)
{
    int w    = (blockIdx.x * blockDim.x + threadIdx.x) >> 5;   // wave id == edge id
    int lane = threadIdx.x & 31;
    if (w >= E + n) return;
    int s, d; edge_endpoints(ei, E, w, s, d);

    float a  = alpha[0];
    float cl = (1.0f - a) * 0.5f;       // (1-alpha) * head-mean
    float ch = a * 0.5f;                // alpha * head-mean

#pragma unroll
    for (int h = 0; h < 2; ++h) {
        float el = lrelu(asl[s * 2 + h] + adl[d * 2 + h]);
        float wl = cl * expf(el - ml[d * 2 + h]) / dl[d * 2 + h];
        float eh = lrelu(ash[s * 2 + h] + adh[d * 2 + h]);
        float wh = ch * expf(eh - mh[d * 2 + h]) / dh[d * 2 + h];

        const float* xl = xpl + (size_t)s * HC + h * COUT;
        const float* xg = xph + (size_t)s * HC + h * COUT;
#pragma unroll
        for (int cc = 0; cc < 2; ++cc) {
            int c = lane + cc * 32;
            atomicAdd(&out[(size_t)d * COUT + c], wl * xl[c] + wh * xg[c]);
        }
    }
}

// ---------------------------------------------------------------------------
extern "C" void kernel_launch(void* const* d_in, const int* in_sizes, int n_in,
                              void* d_out, int out_size, void* d_ws, size_t ws_size,
                              hipStream_t stream)
{
    const float* init_f  = (const float*)d_in[0];
    const float* low_f   = (const float*)d_in[1];
    const int*   ei      = (const int*)d_in[2];
    const float* alpha   = (const float*)d_in[3];
    const float* W_low   = (const float*)d_in[4];
    const float* as_low  = (const float*)d_in[5];
    const float* ad_low  = (const float*)d_in[6];
    const float* b_low   = (const float*)d_in[7];
    const float* W_high  = (const float*)d_in[8];
    const float* as_high = (const float*)d_in[9];
    const float* ad_high = (const float*)d_in[10];
    const float* b_high  = (const float*)d_in[11];
    float* out = (float*)d_out;

    const int n = in_sizes[0] / F_IN;       // 50000
    const int E = in_sizes[2] / 2;          // 800000
    const int ET = E + n;                   // + self loops

    // workspace carve-up (floats): ~n*336 floats (~67 MB)
    float* ws = (float*)d_ws;
    size_t o = 0;
    float* xh   = ws + o; o += (size_t)n * F_IN;   // high-freq features
    float* xpl  = ws + o; o += (size_t)n * HC;     // projected low
    float* xph  = ws + o; o += (size_t)n * HC;     // projected high
    float* aslw = ws + o; o += (size_t)n * 2;      // attn scores
    float* adlw = ws + o; o += (size_t)n * 2;
    float* ashw = ws + o; o += (size_t)n * 2;
    float* adhw = ws + o; o += (size_t)n * 2;
    float* ml   = ws + o; o += (size_t)n * 2;      // segment max
    float* mh   = ws + o; o += (size_t)n * 2;
    float* dl   = ws + o; o += (size_t)n * 2;      // segment denom
    float* dh   = ws + o; o += (size_t)n * 2;

    // 1. prep: high features, init m/-inf, denom/0, out = blended bias
    {
        int threads = n * F_IN;
        prep_kernel<<<(threads + 255) / 256, 256, 0, stream>>>(
            init_f, low_f, xh, ml, mh, dl, dh, b_low, b_high, alpha, out, n);
    }

    // 2. WMMA projections
    {
        int blocks = (n + 15) / 16;
        gemm_xp_kernel<<<blocks, 256, 0, stream>>>(low_f, W_low, xpl, n);
        gemm_xp_kernel<<<blocks, 256, 0, stream>>>(xh,    W_high, xph, n);
    }

    // 3. attention scores
    {
        int threads = n * 2;
        attn_score_kernel<<<(threads + 255) / 256, 256, 0, stream>>>(
            xpl, as_low, ad_low, aslw, adlw, n);
        attn_score_kernel<<<(threads + 255) / 256, 256, 0, stream>>>(
            xph, as_high, ad_high, ashw, adhw, n);
    }

    // 4-6. edge passes
    {
        int blk1 = (ET + 255) / 256;
        edge_max_kernel<<<blk1, 256, 0, stream>>>(ei, E, n, aslw, adlw, ashw, adhw, ml, mh);
        edge_sum_kernel<<<blk1, 256, 0, stream>>>(ei, E, n, aslw, adlw, ashw, adhw,
                                                  ml, mh, dl, dh);
        long long threads = (long long)ET * 32;
        int blk3 = (int)((threads + 255) / 256);
        edge_scatter_kernel<<<blk3, 256, 0, stream>>>(ei, E, n, xpl, xph,
                                                      aslw, adlw, ashw, adhw,
                                                      ml, mh, dl, dh, alpha, out);
    }
}